// non_local_b_57260503990845
// MI455X (gfx1250) — compile-verified
//
#include <hip/hip_runtime.h>

// ---------------------------------------------------------------------------
// Non-local block, algebraically collapsed:
//   G = feat @ feat^T            [256 x 256], K = 10368   (WMMA f32, K-split 6)
//   s = rowsum(feat)             [256]
//   M = Wj@G + bj (x) s          [128 x 256]              (WMMA f32)
//   q = bi @ M                   [256]
//   R = M^T @ Wi                 [256 x 256]              (WMMA f32)
//   out = feat + (R@feat + q(x)1) / N                     (WMMA f32)
// All matrix math on v_wmma_f32_16x16x4_f32 (wave32, 16x16 tiles).
// ---------------------------------------------------------------------------

typedef __attribute__((ext_vector_type(2))) float v2f;
typedef __attribute__((ext_vector_type(4))) float v4f;
typedef __attribute__((ext_vector_type(8))) float v8f;

#define C_DIM   256
#define H_DIM   128
#define N_TOK   10368
#define KSPLIT  6
#define CH      64          // K-chunk staged in LDS
#define LSTR    68          // padded LDS row stride (floats) -> conflict-free

// Generic 16x16-output-tile GEMM, one wave32 per tile, K-split via blockIdx.z.
// Logical op: D(z) = A_mat[16 x Kz] * B_mat[Kz x 16]  (+ epilogue)
//   TA=0 : A_mem[m][k] (k contiguous, ld=lda)   TA=1 : A_mem[k][m]
//   TB=0 : B_mem[k][j] (j contiguous, ld=ldb)   TB=1 : B_mem[j][k]
// EPI=0 : D = acc
// EPI=1 : D = acc + u[row]*v[col]
// EPI=2 : D = addbase[row,col] + (acc + u[row]) * scale
template <bool TA, bool TB, int EPI>
__global__ __launch_bounds__(32) void gemm16_wmma(
    const float* __restrict__ A, const float* __restrict__ B,
    float* __restrict__ D, int lda, int ldb, int ldd, int Kz,
    const float* __restrict__ u, const float* __restrict__ v,
    const float* __restrict__ addbase, float scale)
{
  __shared__ float As[16 * LSTR];
  __shared__ float Bs[16 * LSTR];

  const int lane = threadIdx.x;          // 0..31
  const int m0 = blockIdx.x * 16;
  const int n0 = blockIdx.y * 16;
  const int z  = blockIdx.z;
  const int k_begin = z * Kz;
  const int k_end   = k_begin + Kz;

  const int hl = lane >> 4;              // half-wave: 0 or 1
  const int lr = lane & 15;

  v8f acc = {};                          // 16x16 f32 accumulator (8 VGPRs)

  for (int k0 = k_begin; k0 < k_end; k0 += CH) {
    // ---- stage A chunk into As[m][kk] (row-major, padded) ----
    if (!TA) {
      #pragma unroll
      for (int i = 0; i < 8; ++i) {
        int t  = lane + 32 * i;          // 0..255
        int r  = t >> 4;                 // 0..15
        int c4 = (t & 15) * 4;           // 0..60
        v4f val = *(const v4f*)(A + (size_t)(m0 + r) * lda + k0 + c4);
        *(v4f*)(&As[r * LSTR + c4]) = val;
      }
    } else {
      #pragma unroll
      for (int i = 0; i < 8; ++i) {
        int t  = lane + 32 * i;
        int cc = t >> 2;                 // 0..63 (k within chunk)
        int qd = (t & 3) * 4;            // m quad base
        v4f val = *(const v4f*)(A + (size_t)(k0 + cc) * lda + m0 + qd);
        As[(qd + 0) * LSTR + cc] = val.x;
        As[(qd + 1) * LSTR + cc] = val.y;
        As[(qd + 2) * LSTR + cc] = val.z;
        As[(qd + 3) * LSTR + cc] = val.w;
      }
    }
    // ---- stage B chunk into Bs[j][kk] ----
    if (TB) {
      #pragma unroll
      for (int i = 0; i < 8; ++i) {
        int t  = lane + 32 * i;
        int r  = t >> 4;
        int c4 = (t & 15) * 4;
        v4f val = *(const v4f*)(B + (size_t)(n0 + r) * ldb + k0 + c4);
        *(v4f*)(&Bs[r * LSTR + c4]) = val;
      }
    } else {
      #pragma unroll
      for (int i = 0; i < 8; ++i) {
        int t  = lane + 32 * i;
        int cc = t >> 2;
        int qd = (t & 3) * 4;
        v4f val = *(const v4f*)(B + (size_t)(k0 + cc) * ldb + n0 + qd);
        Bs[(qd + 0) * LSTR + cc] = val.x;
        Bs[(qd + 1) * LSTR + cc] = val.y;
        Bs[(qd + 2) * LSTR + cc] = val.z;
        Bs[(qd + 3) * LSTR + cc] = val.w;
      }
    }
    // prefetch next chunk while we compute (global_prefetch_b8)
    if (k0 + CH < k_end) {
      if (!TA) __builtin_prefetch(A + (size_t)(m0 + lr) * lda + k0 + CH, 0, 0);
      else     __builtin_prefetch(A + (size_t)(k0 + CH + lane * 2) * lda + m0, 0, 0);
      if (TB)  __builtin_prefetch(B + (size_t)(n0 + lr) * ldb + k0 + CH, 0, 0);
      else     __builtin_prefetch(B + (size_t)(k0 + CH + lane * 2) * ldb + n0, 0, 0);
    }
    __syncthreads();

    // ---- 16 WMMAs: K advances 4 per v_wmma_f32_16x16x4_f32 ----
    // A frag (16x4 f32): v0 = K {0|2}, v1 = K {1|3} per half-wave.
    // B frag mirrors A (B^T staged as [j][k]).
    #pragma unroll
    for (int kk = 0; kk < CH; kk += 4) {
      int o = kk + 2 * hl;
      v2f a, b;
      a.x = As[lr * LSTR + o];
      a.y = As[lr * LSTR + o + 1];
      b.x = Bs[lr * LSTR + o];
      b.y = Bs[lr * LSTR + o + 1];
      acc = __builtin_amdgcn_wmma_f32_16x16x4_f32(
          /*neg_a=*/false, a, /*neg_b=*/false, b,
          /*c_mod=*/(short)0, acc, /*reuse_a=*/false, /*reuse_b=*/false);
    }
    __syncthreads();
  }

  // ---- epilogue: C/D layout = lanes 0-15 rows r, lanes 16-31 rows r+8 ----
  float* Dz = D + (size_t)z * (size_t)gridDim.x * 16 * ldd;
  #pragma unroll
  for (int r = 0; r < 8; ++r) {
    int row = m0 + r + 8 * hl;
    int col = n0 + lr;
    float val = acc[r];
    if (EPI == 1) val += u[row] * v[col];
    if (EPI == 2) val = addbase[(size_t)row * ldd + col] + (val + u[row]) * scale;
    Dz[(size_t)row * ldd + col] = val;
  }
}

// s[c] = sum_n feat[c, n]
__global__ __launch_bounds__(256) void rowsum_kernel(
    const float* __restrict__ feat, float* __restrict__ s)
{
  __shared__ float red[256];
  const int c = blockIdx.x, t = threadIdx.x;
  float p = 0.f;
  for (int n = t; n < N_TOK; n += 256) p += feat[(size_t)c * N_TOK + n];
  red[t] = p;
  __syncthreads();
  for (int off = 128; off > 0; off >>= 1) {
    if (t < off) red[t] += red[t + off];
    __syncthreads();
  }
  if (t == 0) s[c] = red[0];
}

// G = sum_z Gp[z]  (deterministic K-split reduction, 256*256 elems)
__global__ __launch_bounds__(256) void gsum_kernel(
    const float* __restrict__ Gp, float* __restrict__ G)
{
  const int i = blockIdx.x * 256 + threadIdx.x;
  float acc = 0.f;
  #pragma unroll
  for (int zz = 0; zz < KSPLIT; ++zz) acc += Gp[zz * (C_DIM * C_DIM) + i];
  G[i] = acc;
}

// q[c] = sum_r bi[r] * M[r, c]
__global__ __launch_bounds__(256) void qvec_kernel(
    const float* __restrict__ bi, const float* __restrict__ M2,
    float* __restrict__ q)
{
  const int c = threadIdx.x;
  float acc = 0.f;
  for (int r = 0; r < H_DIM; ++r) acc += bi[r] * M2[r * C_DIM + c];
  q[c] = acc;
}

extern "C" void kernel_launch(void* const* d_in, const int* in_sizes, int n_in,
                              void* d_out, int out_size, void* d_ws, size_t ws_size,
                              hipStream_t stream) {
  const float* feat = (const float*)d_in[0];   // [256][10368]
  const float* Wi   = (const float*)d_in[1];   // [128][256]
  const float* bi   = (const float*)d_in[2];   // [128]
  const float* Wj   = (const float*)d_in[3];   // [128][256]
  const float* bj   = (const float*)d_in[4];   // [128]
  float* out = (float*)d_out;                  // [256][10368]

  float* ws = (float*)d_ws;
  float* s  = ws;                              // 256
  float* Gp = s + 256;                         // KSPLIT * 256*256
  float* G  = Gp + KSPLIT * C_DIM * C_DIM;     // 256*256
  float* M2 = G + C_DIM * C_DIM;               // 128*256
  float* R  = M2 + H_DIM * C_DIM;              // 256*256
  float* q  = R + C_DIM * C_DIM;               // 256

  // s = rowsum(feat)
  rowsum_kernel<<<C_DIM, 256, 0, stream>>>(feat, s);

  // Gp[z] = feat[:, zK:(z+1)K] @ feat[:, zK:(z+1)K]^T   (A normal, B "j-major")
  gemm16_wmma<false, true, 0><<<dim3(16, 16, KSPLIT), 32, 0, stream>>>(
      feat, feat, Gp, N_TOK, N_TOK, C_DIM, N_TOK / KSPLIT,
      nullptr, nullptr, nullptr, 0.f);
  gsum_kernel<<<C_DIM * C_DIM / 256, 256, 0, stream>>>(Gp, G);

  // M2 = Wj @ G + bj (x) s
  gemm16_wmma<false, false, 1><<<dim3(H_DIM / 16, 16, 1), 32, 0, stream>>>(
      Wj, G, M2, C_DIM, C_DIM, C_DIM, C_DIM, bj, s, nullptr, 0.f);

  // q = bi @ M2
  qvec_kernel<<<1, 256, 0, stream>>>(bi, M2, q);

  // R = M2^T @ Wi   (A transposed in memory, B row-major)
  gemm16_wmma<true, false, 0><<<dim3(16, H_DIM / 16, 1), 32, 0, stream>>>(
      M2, Wi, R, C_DIM, C_DIM, C_DIM, H_DIM,
      nullptr, nullptr, nullptr, 0.f);

  // out = feat + (R @ feat + q (x) 1) / N
  gemm16_wmma<false, false, 2><<<dim3(16, N_TOK / 16, 1), 32, 0, stream>>>(
      R, feat, out, C_DIM, N_TOK, N_TOK, C_DIM,
      q, nullptr, feat, 1.0f / (float)N_TOK);
}